// PW_58351425683468
// MI455X (gfx1250) — compile-verified
//
#include <hip/hip_runtime.h>
#include <hip/hip_bf16.h>

// Problem constants (match reference)
#define BATCH 16
#define NPTS  2048
#define PDIM  3
#define NPROJ 200

#define NTILES (NPTS / 16)          // 128
#define LTILES ((NPROJ + 15) / 16)  // 13
#define WAVES_PER_BLOCK 8

typedef __attribute__((ext_vector_type(2))) float v2f;
typedef __attribute__((ext_vector_type(8))) float v8f;

// ---------------------------------------------------------------------------
// Kernel 1: projection GEMM via V_WMMA_F32_16X16X4_F32.
//   xproj[b,l,n] = sum_d x[b,n,d] * proj[b,l,d]   (K = 3, padded to 4)
// One wave per 16(n) x 16(l) tile; two WMMAs per wave (x and y share B tile).
// A layout (16x4 f32): lanes 0-15 -> {K0,K1}, lanes 16-31 -> {K2,K3(=0)}.
// C/D layout: VGPR r -> row r (lanes 0-15), row r+8 (lanes 16-31).
// ---------------------------------------------------------------------------
__global__ __launch_bounds__(WAVES_PER_BLOCK * 32)
void sw_proj_wmma_kernel(const float* __restrict__ x,
                         const float* __restrict__ y,
                         const float* __restrict__ proj,
                         float* __restrict__ xproj,
                         float* __restrict__ yproj)
{
    const unsigned wave = blockIdx.x * WAVES_PER_BLOCK + (threadIdx.x >> 5);
    const unsigned lane = threadIdx.x & 31u;
    const unsigned tiles_per_batch = NTILES * LTILES;
    if (wave >= BATCH * tiles_per_batch) return;

    const unsigned b   = wave / tiles_per_batch;
    const unsigned rem = wave % tiles_per_batch;
    const unsigned nt  = rem / LTILES;      // point tile
    const unsigned lt  = rem % LTILES;      // projection tile
    const unsigned n0  = nt * 16;
    const unsigned l0  = lt * 16;

    const unsigned m  = lane & 15u;         // row within half-wave
    const bool     hi = lane >= 16u;

    // --- A tiles for x and y -------------------------------------------------
    const float* xb = x + ((size_t)b * NPTS + n0) * PDIM;
    const float* yb = y + ((size_t)b * NPTS + n0) * PDIM;
    v2f ax, ay;
    ax.x = hi ? xb[m * PDIM + 2] : xb[m * PDIM + 0];   // K=2 : K=0
    ax.y = hi ? 0.0f             : xb[m * PDIM + 1];   // K=3(pad) : K=1
    ay.x = hi ? yb[m * PDIM + 2] : yb[m * PDIM + 0];
    ay.y = hi ? 0.0f             : yb[m * PDIM + 1];

    // --- B tile from projections (4x16, col = projection index) -------------
    const unsigned l = l0 + m;
    v2f bm;
    bm.x = 0.0f; bm.y = 0.0f;
    if (l < NPROJ) {
        const float* pb = proj + ((size_t)b * NPROJ + l) * PDIM;
        bm.x = hi ? pb[2] : pb[0];          // K=2 : K=0
        bm.y = hi ? 0.0f  : pb[1];          // K=3(pad) : K=1
    }

    v8f cx = {};
    v8f cy = {};
    cx = __builtin_amdgcn_wmma_f32_16x16x4_f32(false, ax, false, bm,
                                               (short)0, cx, false, false);
    cy = __builtin_amdgcn_wmma_f32_16x16x4_f32(false, ay, false, bm,
                                               (short)0, cy, false, false);

    // --- store D: element r -> (n = n0 + r + hi*8, l = l0 + (lane&15)) ------
    if (l < NPROJ) {
        const size_t base = ((size_t)b * NPROJ + l) * NPTS + n0 + (hi ? 8u : 0u);
        float* xo = xproj + base;
        float* yo = yproj + base;
        #pragma unroll
        for (int r = 0; r < 8; ++r) {
            xo[r] = cx[r];
            yo[r] = cy[r];
        }
    }
}

// ---------------------------------------------------------------------------
// Kernel 2: per-(b,l) bitonic argsort of 2048 keys in LDS (x and y), then
// gather 3-D points in sorted order, squared-diff, block reduce, pre-scaled
// global atomic accumulate.
//
// Each element is packed into one u64: { order_preserving(f32 key), index }.
// - single ds_load_b64 / ds_store_b64 per side of a compare-exchange
//   (half the LDS traffic of separate key/idx arrays)
// - ties broken by ascending index == jnp.argsort stable order
// ---------------------------------------------------------------------------
#define SORT_THREADS 256

__device__ inline unsigned sw_key_encode(float f)
{
    unsigned u = __float_as_uint(f);
    // monotone map: ascending float order -> ascending unsigned order
    return (u & 0x80000000u) ? ~u : (u | 0x80000000u);
}

__device__ inline void sw_bitonic_sort_2048(unsigned long long* skv)
{
    for (unsigned k = 2; k <= NPTS; k <<= 1) {
        for (unsigned j = k >> 1; j > 0; j >>= 1) {
            __syncthreads();
            for (unsigned i = threadIdx.x; i < NPTS; i += SORT_THREADS) {
                const unsigned ixj = i ^ j;
                if (ixj > i) {
                    const unsigned long long a = skv[i];
                    const unsigned long long c = skv[ixj];
                    const bool up = ((i & k) == 0);
                    // ascending: swap if a > c ; descending: swap if a < c
                    if ((a > c) == up) {
                        skv[i]   = c;
                        skv[ixj] = a;
                    }
                }
            }
        }
    }
    __syncthreads();
}

__global__ __launch_bounds__(SORT_THREADS)
void sw_sort_diff_kernel(const float* __restrict__ x,
                         const float* __restrict__ y,
                         const float* __restrict__ xproj,
                         const float* __restrict__ yproj,
                         float* __restrict__ out,
                         float scale)
{
    __shared__ unsigned long long skv[NPTS];   // packed {key, idx}
    __shared__ int   xorder[NPTS];             // saved x argsort result
    __shared__ float red[SORT_THREADS];

    const unsigned bl = blockIdx.x;           // b * NPROJ + l
    const unsigned b  = bl / NPROJ;
    const size_t   row = (size_t)bl * NPTS;

    // ---- argsort x projections ---------------------------------------------
    for (unsigned i = threadIdx.x; i < NPTS; i += SORT_THREADS)
        skv[i] = ((unsigned long long)sw_key_encode(xproj[row + i]) << 32) | i;
    sw_bitonic_sort_2048(skv);
    for (unsigned i = threadIdx.x; i < NPTS; i += SORT_THREADS)
        xorder[i] = (int)(unsigned)(skv[i] & 0xFFFFFFFFu);
    __syncthreads();

    // ---- argsort y projections ---------------------------------------------
    for (unsigned i = threadIdx.x; i < NPTS; i += SORT_THREADS)
        skv[i] = ((unsigned long long)sw_key_encode(yproj[row + i]) << 32) | i;
    sw_bitonic_sort_2048(skv);

    // ---- gather + squared diff ---------------------------------------------
    const float* xb = x + (size_t)b * NPTS * PDIM;
    const float* yb = y + (size_t)b * NPTS * PDIM;
    float acc = 0.0f;
    for (unsigned i = threadIdx.x; i < NPTS; i += SORT_THREADS) {
        const int xi = xorder[i];
        const int yi = (int)(unsigned)(skv[i] & 0xFFFFFFFFu);
        const float d0 = xb[xi * PDIM + 0] - yb[yi * PDIM + 0];
        const float d1 = xb[xi * PDIM + 1] - yb[yi * PDIM + 1];
        const float d2 = xb[xi * PDIM + 2] - yb[yi * PDIM + 2];
        acc += d0 * d0 + d1 * d1 + d2 * d2;
    }

    red[threadIdx.x] = acc;
    __syncthreads();
    for (unsigned s = SORT_THREADS / 2; s > 0; s >>= 1) {
        if (threadIdx.x < s) red[threadIdx.x] += red[threadIdx.x + s];
        __syncthreads();
    }
    if (threadIdx.x == 0)
        atomicAdd(out, red[0] * scale);
}

// ---------------------------------------------------------------------------
__global__ void sw_zero_out_kernel(float* out)
{
    if (threadIdx.x == 0 && blockIdx.x == 0) out[0] = 0.0f;
}

extern "C" void kernel_launch(void* const* d_in, const int* in_sizes, int n_in,
                              void* d_out, int out_size, void* d_ws, size_t ws_size,
                              hipStream_t stream)
{
    const float* x    = (const float*)d_in[0];   // [B, N, D]
    const float* y    = (const float*)d_in[1];   // [B, N, D]
    const float* proj = (const float*)d_in[2];   // [B, L, D]
    float* out = (float*)d_out;

    float* xproj = (float*)d_ws;                                  // [B, L, N]
    float* yproj = xproj + (size_t)BATCH * NPROJ * NPTS;          // [B, L, N]

    sw_zero_out_kernel<<<1, 32, 0, stream>>>(out);

    const unsigned total_waves = BATCH * NTILES * LTILES;         // 26624
    const unsigned proj_blocks = (total_waves + WAVES_PER_BLOCK - 1) / WAVES_PER_BLOCK;
    sw_proj_wmma_kernel<<<proj_blocks, WAVES_PER_BLOCK * 32, 0, stream>>>(
        x, y, proj, xproj, yproj);

    const float scale = 1.0f / ((float)BATCH * NPROJ * NPTS * PDIM);
    sw_sort_diff_kernel<<<BATCH * NPROJ, SORT_THREADS, 0, stream>>>(
        x, y, xproj, yproj, out, scale);
}